// MAF_78134045048834
// MI455X (gfx1250) — compile-verified
//
#include <hip/hip_runtime.h>
#include <hip/hip_bf16.h>
#include <stdint.h>

// MAF forward for MI455X / gfx1250.
// Compute-bound (~567 GFLOP fp32 vs ~1.5 GB HBM traffic) -> fp32 WMMA path
// (v_wmma_f32_16x16x4_f32) for fp32 fidelity on the matrix pipe, with
// GLOBAL_LOAD_ASYNC_TO_LDS_B128 double-buffered tile staging (ASYNCcnt).

#define D_   784
#define H_   1024
#define L_   5
#define B_   16384
#define EPS_ 1e-5f

typedef __attribute__((ext_vector_type(2))) float v2f;
typedef __attribute__((ext_vector_type(8))) float v8f;
typedef int v4i __attribute__((vector_size(16)));   // matches builtin param pointee
typedef __attribute__((address_space(1))) v4i* gptr_v4i;
typedef __attribute__((address_space(3))) v4i* lptr_v4i;

// GEMM tiling
#define BM 128      // block tile M
#define BN 64       // block tile N
#define BK 16       // K step per LDS tile
#define LDA 20      // LDS stride (floats) for A tile (bank-conflict-free frag reads)
#define LDB 72      // LDS stride (floats) for B tile (half-waves hit disjoint banks)
#define N3PAD 1600  // padded width of masked W2 so B-tile loads are never OOB

#if defined(__has_builtin)
#if __has_builtin(__builtin_amdgcn_global_load_async_to_lds_b128) && \
    __has_builtin(__builtin_amdgcn_s_wait_asynccnt)
#define USE_ASYNC 1
#endif
#endif
#ifndef USE_ASYNC
#define USE_ASYNC 0
#endif

__device__ __forceinline__ v8f wmma16x16x4(v2f a, v2f b, v8f c) {
  // (neg_a, A, neg_b, B, c_mod, C, reuse_a, reuse_b)
  return __builtin_amdgcn_wmma_f32_16x16x4_f32(false, a, false, b, (short)0, c,
                                               false, false);
}

#if USE_ASYNC
__device__ __forceinline__ void async_copy16(const float* g, float* l) {
  // 16B per lane, global -> LDS, tracked by ASYNCcnt
  __builtin_amdgcn_global_load_async_to_lds_b128(
      (gptr_v4i)(__attribute__((address_space(1))) void*)(void*)g,
      (lptr_v4i)(__attribute__((address_space(3))) void*)l, 0, 0);
}
#endif

// Stage one (BM x BK) A tile and one (BK x BN) B tile into LDS.
// Exactly 3 async instructions per wave when USE_ASYNC (2x A b128 + 1x B b128).
__device__ __forceinline__ void stage_tile(const float* __restrict__ A,
                                           const float* __restrict__ Bw,
                                           int m0, int n0, int K, int Nb, int k0,
                                           float* la, float* lb, int tid) {
#if USE_ASYNC
  for (int q = tid; q < (BM * BK) / 4; q += 256) {   // 2 iterations
    const int row = q >> 2;                          // 0..127
    const int c4  = (q & 3) << 2;                    // 0,4,8,12
    async_copy16(A + (size_t)(m0 + row) * K + k0 + c4, la + row * LDA + c4);
  }
  {
    const int krow = tid >> 4;                       // 0..15
    const int c4   = (tid & 15) << 2;                // 0..60
    async_copy16(Bw + (size_t)(k0 + krow) * Nb + n0 + c4, lb + krow * LDB + c4);
  }
#else
  for (int q = tid; q < (BM * BK) / 4; q += 256) {
    const int row = q >> 2;
    const int c4  = (q & 3) << 2;
    const float4 v = *(const float4*)(A + (size_t)(m0 + row) * K + k0 + c4);
    *(float4*)(la + row * LDA + c4) = v;
  }
  {
    const int krow = tid >> 4;
    const int c4   = (tid & 15) << 2;
    const float4 v = *(const float4*)(Bw + (size_t)(k0 + krow) * Nb + n0 + c4);
    *(float4*)(lb + krow * LDB + c4) = v;
  }
#endif
}

// C[M,N] = act(A[M,K] @ Bw[K,Nb](cols 0..N-1) + bias[N]); row-major fp32.
// M % 128 == 0, K % 16 == 0, Nb >= gridDim.x*BN (B loads never OOB).
__global__ __launch_bounds__(256) void maf_gemm(
    const float* __restrict__ A, const float* __restrict__ Bw,
    const float* __restrict__ bias, float* __restrict__ C,
    int M, int N, int Nb, int K, int relu)
{
  __shared__ float lds_a[2 * BM * LDA];  // 20.0 KB (double buffered)
  __shared__ float lds_b[2 * BK * LDB];  //  9.0 KB

  const int tid  = threadIdx.x;
  const int lane = tid & 31;
  const int wave = tid >> 5;     // 0..7
  const int wm   = wave & 3;     // 4 waves tile M (32 rows each)
  const int wn   = wave >> 2;    // 2 waves tile N (32 cols each)
  const int half = lane >> 4;    // 0: lanes 0-15, 1: lanes 16-31
  const int l16  = lane & 15;

  const int m0 = blockIdx.y * BM;
  const int n0 = blockIdx.x * BN;

  v8f acc[2][2];
  #pragma unroll
  for (int mi = 0; mi < 2; ++mi)
    #pragma unroll
    for (int ni = 0; ni < 2; ++ni)
      #pragma unroll
      for (int e = 0; e < 8; ++e)
        acc[mi][ni][e] = 0.f;

  const int a_row = wm * 32 + l16;   // local A row for this lane (+16*mi)
  const int b_col = wn * 32 + l16;   // local B col for this lane (+16*ni)

  const int nt = K / BK;
  stage_tile(A, Bw, m0, n0, K, Nb, 0, lds_a, lds_b, tid);

  for (int t = 0; t < nt; ++t) {
    const int cur = t & 1;
    float* la = lds_a + cur * (BM * LDA);
    float* lb = lds_b + cur * (BK * LDB);

    if (t + 1 < nt) {
      // prefetch next tile into the other buffer, then wait for current tile
      stage_tile(A, Bw, m0, n0, K, Nb, (t + 1) * BK,
                 lds_a + (cur ^ 1) * (BM * LDA),
                 lds_b + (cur ^ 1) * (BK * LDB), tid);
#if USE_ASYNC
      __builtin_amdgcn_s_wait_asynccnt(3);   // allow the 3 next-tile ops in flight
#endif
    } else {
#if USE_ASYNC
      __builtin_amdgcn_s_wait_asynccnt(0);
#endif
    }
    __syncthreads();

    // ---- 4 ksteps x (2x2) v_wmma_f32_16x16x4_f32 on the current tile ----
    #pragma unroll
    for (int k = 0; k < BK; k += 4) {
      const int kk = k + 2 * half;   // A/B frag: lanes<16 -> K=k,k+1 ; lanes>=16 -> K=k+2,k+3
      v2f afrag[2], bfrag[2];
      #pragma unroll
      for (int mi = 0; mi < 2; ++mi)
        afrag[mi] = *(const v2f*)(la + (a_row + 16 * mi) * LDA + kk);
      #pragma unroll
      for (int ni = 0; ni < 2; ++ni) {
        const float* p = lb + kk * LDB + b_col + 16 * ni;
        v2f b; b.x = p[0]; b.y = p[LDB];
        bfrag[ni] = b;
      }
      #pragma unroll
      for (int mi = 0; mi < 2; ++mi)
        #pragma unroll
        for (int ni = 0; ni < 2; ++ni)
          acc[mi][ni] = wmma16x16x4(afrag[mi], bfrag[ni], acc[mi][ni]);
    }
    __syncthreads();
  }

  // ---- epilogue: bias (+ReLU), guarded stores ----
  // C/D layout: VGPR r -> lanes 0-15: (M=r, N=lane), lanes 16-31: (M=8+r, N=lane-16)
  #pragma unroll
  for (int mi = 0; mi < 2; ++mi) {
    #pragma unroll
    for (int ni = 0; ni < 2; ++ni) {
      const int n = n0 + wn * 32 + ni * 16 + l16;
      const float bv = (n < N) ? bias[n] : 0.f;
      #pragma unroll
      for (int r = 0; r < 8; ++r) {
        const int m = m0 + wm * 32 + mi * 16 + r + 8 * half;
        float v = acc[mi][ni][r] + bv;
        if (relu) v = fmaxf(v, 0.f);
        if (n < N) C[(size_t)m * N + n] = v;
      }
    }
  }
}

// Wm[K,Npad]: cols <N get W*mask (masks are jnp bool -> 1 byte), pad cols get 0.
__global__ __launch_bounds__(256) void maf_maskmul(
    const float* __restrict__ W, const unsigned char* __restrict__ Mk,
    float* __restrict__ Wm, int K, int N, int Npad)
{
  const int total = K * Npad;
  for (int i = blockIdx.x * blockDim.x + threadIdx.x; i < total;
       i += gridDim.x * blockDim.x) {
    const int k = i / Npad;
    const int n = i - k * Npad;
    float v = 0.f;
    if (n < N) {
      const size_t s = (size_t)k * N + n;
      v = Mk[s] ? W[s] : 0.f;
    }
    Wm[i] = v;
  }
}

// Per-row flow update: x' = (x - mu) * exp(-alpha); logdet -= sum(alpha);
// optional running-stat batchnorm + its logdet term.
__global__ __launch_bounds__(256) void maf_flow(
    const float* __restrict__ x_in, const float* __restrict__ out,
    const float* __restrict__ gamma, const float* __restrict__ beta,
    const float* __restrict__ rmean, const float* __restrict__ rvar,
    float* __restrict__ x_out, float* __restrict__ logdet,
    int do_bn, int first)
{
  const int row = blockIdx.x;
  const int tid = threadIdx.x;
  const float* orow  = out  + (size_t)row * (2 * D_);
  const float* xrow  = x_in + (size_t)row * D_;
  float*       xorow = x_out + (size_t)row * D_;

  float asum = 0.f, bsum = 0.f;
  for (int d = tid; d < D_; d += 256) {
    const float mu = orow[d];
    const float al = orow[D_ + d];
    float xv = (xrow[d] - mu) * expf(-al);
    asum += al;
    if (do_bn) {
      const float std = sqrtf(rvar[d] + EPS_);
      xv = gamma[d] * (xv - rmean[d]) / std + beta[d];
      bsum += logf(fabsf(gamma[d] / std));
    }
    xorow[d] = xv;
  }

  __shared__ float s1[256], s2[256];
  s1[tid] = asum; s2[tid] = bsum;
  __syncthreads();
  for (int s = 128; s > 0; s >>= 1) {
    if (tid < s) { s1[tid] += s1[tid + s]; s2[tid] += s2[tid + s]; }
    __syncthreads();
  }
  if (tid == 0) {
    const float ld = -s1[0] + (do_bn ? s2[0] : 0.f);
    logdet[row] = first ? ld : (logdet[row] + ld);
  }
}

extern "C" void kernel_launch(void* const* d_in, const int* in_sizes, int n_in,
                              void* d_out, int out_size, void* d_ws, size_t ws_size,
                              hipStream_t stream) {
  const float* x  = (const float*)d_in[0];
  const float* W0 = (const float*)d_in[1];
  const float* b0 = (const float*)d_in[2];
  const float* W1 = (const float*)d_in[3];
  const float* b1 = (const float*)d_in[4];
  const float* W2 = (const float*)d_in[5];
  const float* b2 = (const float*)d_in[6];
  const unsigned char* m0 = (const unsigned char*)d_in[7];
  const unsigned char* m1 = (const unsigned char*)d_in[8];
  const unsigned char* m2 = (const unsigned char*)d_in[9];
  const float* gamma = (const float*)d_in[10];
  const float* beta  = (const float*)d_in[11];
  const float* rmean = (const float*)d_in[12];
  const float* rvar  = (const float*)d_in[13];

  float* outx   = (float*)d_out;               // [B, D]
  float* logdet = outx + (size_t)B_ * D_;      // [B]

  // workspace layout (~354 MB): x ping-pong, h0/h1, out, masked-weight staging
  float* ws = (float*)d_ws;
  size_t off = 0;
  float* xA  = ws + off; off += (size_t)B_ * D_;
  float* xB  = ws + off; off += (size_t)B_ * D_;
  float* h0  = ws + off; off += (size_t)B_ * H_;
  float* h1  = ws + off; off += (size_t)B_ * H_;
  float* ob  = ws + off; off += (size_t)B_ * 2 * D_;
  float* Wm0 = ws + off; off += (size_t)D_ * H_;
  float* Wm1 = ws + off; off += (size_t)H_ * H_;
  float* Wm2 = ws + off; off += (size_t)H_ * N3PAD;   // padded to 1600 cols
  (void)ws_size; (void)in_sizes; (void)n_in; (void)out_size;

  const dim3 blk(256);
  for (int i = 0; i < L_; ++i) {
    maf_maskmul<<<512, blk, 0, stream>>>(W0 + (size_t)i * D_ * H_,
                                         m0 + (size_t)i * D_ * H_,
                                         Wm0, D_, H_, H_);
    maf_maskmul<<<512, blk, 0, stream>>>(W1 + (size_t)i * H_ * H_,
                                         m1 + (size_t)i * H_ * H_,
                                         Wm1, H_, H_, H_);
    maf_maskmul<<<512, blk, 0, stream>>>(W2 + (size_t)i * H_ * 2 * D_,
                                         m2 + (size_t)i * H_ * 2 * D_,
                                         Wm2, H_, 2 * D_, N3PAD);

    const float* xin  = (i == 0)      ? x    : ((i & 1) ? xA : xB);
    float*       xout = (i == L_ - 1) ? outx : ((i & 1) ? xB : xA);

    maf_gemm<<<dim3(H_ / BN, B_ / BM), blk, 0, stream>>>(
        xin, Wm0, b0 + (size_t)i * H_, h0, B_, H_, H_, D_, 1);
    maf_gemm<<<dim3(H_ / BN, B_ / BM), blk, 0, stream>>>(
        h0, Wm1, b1 + (size_t)i * H_, h1, B_, H_, H_, H_, 1);
    maf_gemm<<<dim3(N3PAD / BN, B_ / BM), blk, 0, stream>>>(
        h1, Wm2, b2 + (size_t)i * 2 * D_, ob, B_, 2 * D_, N3PAD, H_, 0);

    maf_flow<<<B_, blk, 0, stream>>>(xin, ob,
                                     gamma + (size_t)i * D_, beta + (size_t)i * D_,
                                     rmean + (size_t)i * D_, rvar + (size_t)i * D_,
                                     xout, logdet,
                                     (i < L_ - 1) ? 1 : 0, (i == 0) ? 1 : 0);
  }
}